// DeepGCN_52218212385223
// MI455X (gfx1250) — compile-verified
//
#include <hip/hip_runtime.h>
#include <cstdint>
#include <cstddef>

// DeepGCN (eval) for MI455X / gfx1250.
// GEMMs on the matrix pipe via V_WMMA_F32_16X16X4_F32 (fp32 in/out, matches
// reference precision). Edge aggregation is the roofline bottleneck (~1.3 GB
// of gather + atomic traffic) and is done with float4-per-lane loads and
// global_atomic_add_f32 scatter.

#define DIN 512
#define HID1 128
#define HID2 64
#define NCLS 2
#define BN_EPS 1e-5f

typedef __attribute__((ext_vector_type(2))) float v2f;
typedef __attribute__((ext_vector_type(8))) float v8f;

// ---------------------------------------------------------------- utilities
__global__ void k_fill(float* __restrict__ p, long long n, float v) {
    long long i = (long long)blockIdx.x * blockDim.x + threadIdx.x;
    if (i < n) p[i] = v;
}

__global__ void k_count_deg(const long long* __restrict__ dst, long long E,
                            float* __restrict__ deg) {
    long long e = (long long)blockIdx.x * blockDim.x + threadIdx.x;
    if (e < E) atomicAdd(&deg[(int)dst[e]], 1.0f);
}

__global__ void k_rsqrt_inplace(float* __restrict__ deg, int n) {
    int i = blockIdx.x * blockDim.x + threadIdx.x;
    if (i < n) deg[i] = rsqrtf(deg[i]);   // deg >= 1 always (self loop)
}

// ----------------------------------------------------------- fp32 WMMA GEMM
// H[M x NOUT] = A[M x K] * W[K x NOUT].  One wave (32 lanes) per 16-row
// stripe; NT = NOUT/16 accumulator tiles so A is streamed exactly once.
// Fragment layouts per CDNA5 ISA 7.12.2:
//   A (16x4 f32):  lanes 0-15 hold {K, K+1}, lanes 16-31 hold {K+2, K+3}
//   B (4x16 f32):  VGPR0 = row K (lanes 0-15) / row K+2 (lanes 16-31), etc.
//   C/D (16x16):   VGPR r = row r (lanes 0-15) / row r+8 (lanes 16-31)
// Requires M % 16 == 0 (N = 50000 = 3125*16) so EXEC is all ones for WMMA.
template <int K, int NOUT>
__global__ __launch_bounds__(32) void k_gemm_wmma(const float* __restrict__ A,
                                                  const float* __restrict__ W,
                                                  float* __restrict__ H) {
    constexpr int NT = NOUT / 16;
    const int lane = threadIdx.x;
    const int half = lane >> 4;      // 0: lanes 0-15, 1: lanes 16-31
    const int l    = lane & 15;
    const int row  = blockIdx.x * 16 + l;

    v8f acc[NT];
#pragma unroll
    for (int t = 0; t < NT; ++t) acc[t] = {};

    const float* arow = A + (size_t)row * K + half * 2;
    for (int k = 0; k < K; k += 4) {
        v2f a;
        a.x = arow[k];
        a.y = arow[k + 1];
        const float* wrow = W + (size_t)(k + half * 2) * NOUT + l;
#pragma unroll
        for (int t = 0; t < NT; ++t) {
            v2f b;
            b.x = wrow[t * 16];          // B row (k + half*2)
            b.y = wrow[NOUT + t * 16];   // B row (k + half*2 + 1)
            acc[t] = __builtin_amdgcn_wmma_f32_16x16x4_f32(
                false, a, false, b, (short)0, acc[t], false, false);
        }
    }

    const int orow = blockIdx.x * 16 + half * 8;
#pragma unroll
    for (int t = 0; t < NT; ++t)
#pragma unroll
        for (int r = 0; r < 8; ++r)
            H[(size_t)(orow + r) * NOUT + t * 16 + l] = acc[t][r];
}

// ----------------------------------------------------- edge scatter (F wide)
// One wave per (edge or self-loop).  F=128: float4 per lane; F=64: float2.
template <int F>
__global__ __launch_bounds__(256) void k_scatter_edges(
    const float* __restrict__ h, const long long* __restrict__ src,
    const long long* __restrict__ dst, long long E, int N,
    const float* __restrict__ dis, float* __restrict__ out) {
    const int lane = threadIdx.x & 31;
    const long long e = (long long)blockIdx.x * 8 + (threadIdx.x >> 5);
    const long long total = E + (long long)N;
    if (e >= total) return;
    int s, d;
    if (e < E) { s = (int)src[e]; d = (int)dst[e]; }
    else       { s = d = (int)(e - E); }           // self loop
    const float norm = dis[s] * dis[d];
    if (F == 128) {
        const float4 v =
            reinterpret_cast<const float4*>(h + (size_t)s * F)[lane];
        float* o = out + (size_t)d * F + lane * 4;
        atomicAdd(o + 0, v.x * norm);
        atomicAdd(o + 1, v.y * norm);
        atomicAdd(o + 2, v.z * norm);
        atomicAdd(o + 3, v.w * norm);
    } else {  // F == 64
        const float2 v =
            reinterpret_cast<const float2*>(h + (size_t)s * F)[lane];
        float* o = out + (size_t)d * F + lane * 2;
        atomicAdd(o + 0, v.x * norm);
        atomicAdd(o + 1, v.y * norm);
    }
}

__global__ void k_scatter_c2(const float* __restrict__ h,
                             const long long* __restrict__ src,
                             const long long* __restrict__ dst, long long E,
                             int N, const float* __restrict__ dis,
                             float* __restrict__ out) {
    long long e = (long long)blockIdx.x * blockDim.x + threadIdx.x;
    long long total = E + (long long)N;
    if (e >= total) return;
    int s, d;
    if (e < E) { s = (int)src[e]; d = (int)dst[e]; }
    else       { s = d = (int)(e - E); }
    float norm = dis[s] * dis[d];
    atomicAdd(out + (size_t)d * 2 + 0, h[(size_t)s * 2 + 0] * norm);
    atomicAdd(out + (size_t)d * 2 + 1, h[(size_t)s * 2 + 1] * norm);
}

// --------------------------------------------------- bias + batchnorm + relu
template <int F>
__global__ void k_bn_relu(const float* __restrict__ agg,
                          const float* __restrict__ bias,
                          const float* __restrict__ g,
                          const float* __restrict__ beta,
                          const float* __restrict__ mean,
                          const float* __restrict__ var,
                          float* __restrict__ out, long long total) {
    long long i = (long long)blockIdx.x * blockDim.x + threadIdx.x;
    if (i >= total) return;
    int f = (int)(i & (long long)(F - 1));   // F is a power of two
    float h = agg[i] + bias[f];
    float y = (h - mean[f]) * rsqrtf(var[f] + BN_EPS) * g[f] + beta[f];
    out[i] = fmaxf(y, 0.0f);
}

// ------------------------------------------------ tiny layer-3 GEMM (64 -> 2)
__global__ void k_gemm3(const float* __restrict__ P,
                        const float* __restrict__ W3, float* __restrict__ H3,
                        int N) {
    int i = blockIdx.x * blockDim.x + threadIdx.x;
    if (i >= N) return;
    const float* p = P + (size_t)i * HID2;
    float s0 = 0.f, s1 = 0.f;
#pragma unroll
    for (int k = 0; k < HID2; ++k) {
        float x = p[k];
        s0 += x * W3[k * NCLS + 0];
        s1 += x * W3[k * NCLS + 1];
    }
    H3[(size_t)i * 2 + 0] = s0;
    H3[(size_t)i * 2 + 1] = s1;
}

__global__ void k_bias2(float* __restrict__ out, const float* __restrict__ b3,
                        long long total) {
    long long i = (long long)blockIdx.x * blockDim.x + threadIdx.x;
    if (i < total) out[i] += b3[i & 1];
}

// --------------------------------------------------------------------- host
extern "C" void kernel_launch(void* const* d_in, const int* in_sizes, int n_in,
                              void* d_out, int out_size, void* d_ws,
                              size_t ws_size, hipStream_t stream) {
    const float*     x   = (const float*)d_in[0];
    const long long* ei  = (const long long*)d_in[1];   // int64 [2, E]
    const float*     w1  = (const float*)d_in[2];
    const float*     b1  = (const float*)d_in[3];
    const float*     g1  = (const float*)d_in[4];
    const float*     be1 = (const float*)d_in[5];
    const float*     m1  = (const float*)d_in[6];
    const float*     v1  = (const float*)d_in[7];
    const float*     w2  = (const float*)d_in[8];
    const float*     b2  = (const float*)d_in[9];
    const float*     g2  = (const float*)d_in[10];
    const float*     be2 = (const float*)d_in[11];
    const float*     m2  = (const float*)d_in[12];
    const float*     v2  = (const float*)d_in[13];
    const float*     w3  = (const float*)d_in[14];
    const float*     b3  = (const float*)d_in[15];

    const int       N   = in_sizes[0] / DIN;   // 50000 (divisible by 16)
    const long long E   = in_sizes[1] / 2;     // 800000
    const long long* src = ei;
    const long long* dst = ei + E;

    auto a256 = [](size_t v) { return (v + 255) & ~(size_t)255; };
    char*  ws  = (char*)d_ws;
    float* dis = (float*)ws;                          // N floats (deg -> d^-1/2)
    size_t off = a256((size_t)N * 4);
    size_t szB = a256((size_t)N * HID1 * 4);
    float* B1  = (float*)(ws + off);                  // N x 128 ping
    float* B2  = (float*)(ws + off + szB);            // N x 128 pong
    float* out = (float*)d_out;

    const long long tot = E + (long long)N;
    const unsigned gEdge = (unsigned)((tot + 7) / 8); // 8 waves / 256-thr block

    // degrees on A + I:  deg = 1 (self loop) + count over dst, then rsqrt
    k_fill<<<(N + 255) / 256, 256, 0, stream>>>(dis, N, 1.0f);
    k_count_deg<<<(unsigned)((E + 255) / 256), 256, 0, stream>>>(dst, E, dis);
    k_rsqrt_inplace<<<(N + 255) / 256, 256, 0, stream>>>(dis, N);

    // ---- layer 1: h1 = x @ w1 ; a1 = scatter(norm * h1[src]) ; bn+relu
    k_gemm_wmma<DIN, HID1><<<N / 16, 32, 0, stream>>>(x, w1, B1);
    {
        long long n = (long long)N * HID1;
        k_fill<<<(unsigned)((n + 255) / 256), 256, 0, stream>>>(B2, n, 0.0f);
        k_scatter_edges<HID1><<<gEdge, 256, 0, stream>>>(B1, src, dst, E, N,
                                                         dis, B2);
        k_bn_relu<HID1><<<(unsigned)((n + 255) / 256), 256, 0, stream>>>(
            B2, b1, g1, be1, m1, v1, B1, n);
    }

    // ---- layer 2
    k_gemm_wmma<HID1, HID2><<<N / 16, 32, 0, stream>>>(B1, w2, B2);
    {
        long long n = (long long)N * HID2;
        k_fill<<<(unsigned)((n + 255) / 256), 256, 0, stream>>>(B1, n, 0.0f);
        k_scatter_edges<HID2><<<gEdge, 256, 0, stream>>>(B2, src, dst, E, N,
                                                         dis, B1);
        k_bn_relu<HID2><<<(unsigned)((n + 255) / 256), 256, 0, stream>>>(
            B1, b2, g2, be2, m2, v2, B2, n);
    }

    // ---- layer 3: h3 = p2 @ w3 ; out = scatter + b3
    k_gemm3<<<(N + 255) / 256, 256, 0, stream>>>(B2, w3, B1, N);
    {
        long long n = (long long)N * NCLS;
        k_fill<<<(unsigned)((n + 255) / 256), 256, 0, stream>>>(out, n, 0.0f);
        k_scatter_c2<<<(unsigned)((tot + 255) / 256), 256, 0, stream>>>(
            B1, src, dst, E, N, dis, out);
        k_bias2<<<(unsigned)((n + 255) / 256), 256, 0, stream>>>(out, b3, n);
    }
}